// BiMambaBlock_3770981286790
// MI455X (gfx1250) — compile-verified
//
#include <hip/hip_runtime.h>
#include <hip/hip_bf16.h>
#include <math.h>

typedef __attribute__((ext_vector_type(2))) float v2f;
typedef __attribute__((ext_vector_type(8))) float v8f;

#define DIM    512
#define DINNER 512
#define DSTATE 16
#define DTRANK 32
#define DCONV  4
#define SEQ    4096
#define BATCH  2
#define ROWS   (BATCH * SEQ)   // 8192

#define APAD   36              // A-tile row stride (floats): 16B-aligned, conflict-free
#define BKPAD  34              // Bt row stride (floats): 8B-aligned b64 reads, conflict-free
#define NCHUNK 8
#define CLEN   (SEQ / NCHUNK)  // 512
#define NCH    (2 * BATCH * DINNER)       // 2048 scan channels
#define NTASK  (NCH * NCHUNK)             // 16384 chunk tasks

__device__ __forceinline__ float softplusf(float x) {
  return (x > 20.f) ? x : log1pf(__expf(x));
}

// CDNA5 async global->LDS copy: each lane copies 16B to its own LDS address.
__device__ __forceinline__ void async_copy_b128(void* lds_ptr, const float* gptr) {
  uint32_t lds_off = (uint32_t)(uintptr_t)lds_ptr;   // low 32 bits of generic addr = LDS offset
  asm volatile("global_load_async_to_lds_b128 %0, %1, off"
               :: "v"(lds_off), "v"((uint64_t)(uintptr_t)gptr)
               : "memory");
}
__device__ __forceinline__ void wait_async0() {
  asm volatile("s_wait_asynccnt 0x0" ::: "memory");
}

// ---------------------------------------------------------------------------
// 1) LayerNorm: one wave per row (512 cols = 16 per lane), shuffle reduction
// ---------------------------------------------------------------------------
__global__ void ln_kernel(const float* __restrict__ x,
                          const float* __restrict__ g,
                          const float* __restrict__ b,
                          float* __restrict__ xn) {
  const int row  = blockIdx.x * 8 + (threadIdx.x >> 5);
  const int lane = threadIdx.x & 31;
  const float* xr = x + (size_t)row * DIM;
  float vals[16];
  float s = 0.f, s2 = 0.f;
#pragma unroll
  for (int i = 0; i < 16; ++i) {
    float v = xr[lane + 32 * i];
    vals[i] = v; s += v; s2 += v * v;
  }
#pragma unroll
  for (int off = 16; off > 0; off >>= 1) {
    s  += __shfl_xor(s,  off, 32);
    s2 += __shfl_xor(s2, off, 32);
  }
  const float mu  = s * (1.f / DIM);
  const float var = s2 * (1.f / DIM) - mu * mu;
  const float rs  = rsqrtf(var + 1e-5f);
  float* xo = xn + (size_t)row * DIM;
#pragma unroll
  for (int i = 0; i < 16; ++i) {
    int c = lane + 32 * i;
    xo[c] = (vals[i] - mu) * rs * g[c] + b[c];
  }
}

// ---------------------------------------------------------------------------
// 2) f32 WMMA GEMM. Block = 8 waves; tile 128(M)x64(N); K chunks of 32.
//    A: async global->LDS copies (double-buffered).
//    B: reg-staged + stored TRANSPOSED (Bt[n][k]) so each WMMA B operand is
//       one contiguous ds_load_b64 into an aligned VGPR pair (no v_mov fixup).
//    EPI: 0 = none, 1 = softplus(acc + bias[n])
// ---------------------------------------------------------------------------
template <int EPI>
__global__ void wmma_gemm_f32(const float* __restrict__ A, int lda,
                              const float* __restrict__ B, int ldb,
                              float* __restrict__ C, int ldc,
                              int K, const float* __restrict__ bias) {
  __shared__ float As[2][128 * APAD];     // 2 x 18.4 KB
  __shared__ float Bt[2][64 * BKPAD];     // 2 x  8.7 KB, [n][k]

  const int tid  = threadIdx.x;
  const int wave = tid >> 5;
  const int lane = tid & 31;
  const int m    = lane & 15;
  const int koff = (lane >> 4) * 2;

  const int rowBase = blockIdx.x * 128 + wave * 16;
  const int nBase   = blockIdx.y * 64;

  const int arow = tid >> 3;        // 0..31 (+32/round, 4 rounds -> 128 rows)
  const int acol = (tid & 7) * 4;   // 0..28
  const int brow = tid >> 4;        // k: 0..15 (+16 on round 2)
  const int bcol = (tid & 15) * 4;  // n-local: 0..60

  const float* Ag = A + (size_t)(blockIdx.x * 128 + arow) * lda + acol;
  const float* Bg = B + (size_t)brow * ldb + nBase + bcol;

  auto load_A_async = [&](int buf, int kk) {
    const float* ag = Ag + kk;
#pragma unroll
    for (int r = 0; r < 4; ++r)
      async_copy_b128(&As[buf][(arow + 32 * r) * APAD + acol],
                      ag + (size_t)(32 * r) * lda);
  };

  float4 breg0, breg1;
  auto load_B_global = [&](int kk) {
    breg0 = *(const float4*)(Bg + (size_t)kk * ldb);
    breg1 = *(const float4*)(Bg + (size_t)(kk + 16) * ldb);
  };
  auto store_B_lds = [&](int buf) {
    float* bt = &Bt[buf][0];
    bt[(bcol + 0) * BKPAD + brow] = breg0.x;
    bt[(bcol + 1) * BKPAD + brow] = breg0.y;
    bt[(bcol + 2) * BKPAD + brow] = breg0.z;
    bt[(bcol + 3) * BKPAD + brow] = breg0.w;
    bt[(bcol + 0) * BKPAD + brow + 16] = breg1.x;
    bt[(bcol + 1) * BKPAD + brow + 16] = breg1.y;
    bt[(bcol + 2) * BKPAD + brow + 16] = breg1.z;
    bt[(bcol + 3) * BKPAD + brow + 16] = breg1.w;
  };

  v8f acc[4] = {};

  load_A_async(0, 0);
  load_B_global(0);
  store_B_lds(0);
  wait_async0();
  __syncthreads();

  const int nChunks = K >> 5;
  for (int c = 0; c < nChunks; ++c) {
    const int buf  = c & 1;
    const bool more = (c + 1 < nChunks);
    if (more) {                                   // overlap next chunk with compute
      load_A_async(buf ^ 1, (c + 1) << 5);
      load_B_global((c + 1) << 5);
    }

    const float* Ab = &As[buf][(wave * 16 + m) * APAD];
    const float* Bb = &Bt[buf][m * BKPAD];
#pragma unroll
    for (int k8 = 0; k8 < 8; ++k8) {
      const int k0 = 4 * k8 + koff;
      v2f a = *(const v2f*)(Ab + k0);                     // ds_load_b64
#pragma unroll
      for (int j = 0; j < 4; ++j) {
        v2f bv = *(const v2f*)(Bb + j * (16 * BKPAD) + k0);  // ds_load_b64
        acc[j] = __builtin_amdgcn_wmma_f32_16x16x4_f32(
            /*neg_a=*/false, a, /*neg_b=*/false, bv,
            /*c_mod=*/(short)0, acc[j], /*reuse_a=*/false, /*reuse_b=*/false);
      }
    }
    if (more) store_B_lds(buf ^ 1);
    wait_async0();
    __syncthreads();
  }

  const int rbase = rowBase + 8 * (lane >> 4);
#pragma unroll
  for (int j = 0; j < 4; ++j) {
    const int n = nBase + 16 * j + m;
    const float bv = (EPI == 1) ? bias[n] : 0.f;
#pragma unroll
    for (int r = 0; r < 8; ++r) {
      float v = acc[j][r];
      if (EPI == 1) v = softplusf(v + bv);
      C[(size_t)(rbase + r) * ldc + n] = v;
    }
  }
}

// ---------------------------------------------------------------------------
// 3) Depthwise causal conv (fwd) + anticausal conv (bwd) + softplus.
// ---------------------------------------------------------------------------
__global__ void conv_kernel(const float* __restrict__ zx,
                            const float* __restrict__ cw,   // [512,1,4]
                            const float* __restrict__ cb,   // [512]
                            float* __restrict__ u_f,
                            float* __restrict__ u_b) {
  const int gid = blockIdx.x * blockDim.x + threadIdx.x;   // ROWS*DINNER
  const int d   = gid & (DINNER - 1);
  const int row = gid >> 9;
  const int bb  = row >> 12;
  const int t   = row & (SEQ - 1);

  const float w0 = cw[d * 4 + 0], w1 = cw[d * 4 + 1];
  const float w2 = cw[d * 4 + 2], w3 = cw[d * 4 + 3];
  const float bias = cb[d];
  const float* xin = zx + (size_t)bb * SEQ * 1024 + d;

  float af = bias;
  if (t >= 3) af += w0 * xin[(size_t)(t - 3) * 1024];
  if (t >= 2) af += w1 * xin[(size_t)(t - 2) * 1024];
  if (t >= 1) af += w2 * xin[(size_t)(t - 1) * 1024];
  af += w3 * xin[(size_t)t * 1024];
  float ab = bias;
  if (t + 3 < SEQ) ab += w0 * xin[(size_t)(t + 3) * 1024];
  if (t + 2 < SEQ) ab += w1 * xin[(size_t)(t + 2) * 1024];
  if (t + 1 < SEQ) ab += w2 * xin[(size_t)(t + 1) * 1024];
  ab += w3 * xin[(size_t)t * 1024];

  u_f[(size_t)row * DINNER + d] = softplusf(af);
  u_b[(size_t)row * DINNER + d] = softplusf(ab);
}

// ---------------------------------------------------------------------------
// 4) Chunked selective scan (3 phases).
//    Task = (channel, chunk): channel = dir x batch x d (2048), 8 chunks of 512.
//    Recurrence h = exp(delta*A_n)*h + delta*B_n*u is linear in h, and the
//    chunk decay product is exp(A_n * sum(delta)) -> one running delta-sum.
//    Phase1: local scan (h_in = 0) -> per-chunk (E, L).
//    Phase2: 8-step serial combine per (channel, n) -> per-chunk h_in.
//    Phase3: re-scan with true h_in, emit y (+ u*D), xor-shuffle reduce over n.
// ---------------------------------------------------------------------------
__global__ void scan_part1(const float* __restrict__ delta_f,
                           const float* __restrict__ delta_b,
                           const float* __restrict__ u_f,
                           const float* __restrict__ u_b,
                           const float* __restrict__ dbc_f,   // [8192,64] dt|B|C
                           const float* __restrict__ dbc_b,
                           const float* __restrict__ A_log,
                           float* __restrict__ Lbuf,          // [NTASK,16]
                           float* __restrict__ Ebuf) {        // [NTASK,16]
  const int wave = threadIdx.x >> 5;
  const int lane = threadIdx.x & 31;
  const int n    = lane & 15;
  const int task = blockIdx.x * 16 + wave * 2 + (lane >> 4);  // [0, NTASK)
  const int ck   = task & (NCHUNK - 1);
  const int ch   = task >> 3;
  const int dir  = ch >> 10;
  const int bb   = (ch >> 9) & 1;
  const int d    = ch & (DINNER - 1);

  const float* dlt = dir ? delta_b : delta_f;
  const float* uu  = dir ? u_b : u_f;
  const float* dbc = dir ? dbc_b : dbc_f;

  const float An = -__expf(A_log[d * DSTATE + n]);
  const int base = bb * SEQ;
  const int step = dir ? -1 : 1;
  int t = dir ? (SEQ - 1 - ck * CLEN) : (ck * CLEN);

  size_t pd = (size_t)(base + t) * DINNER + d;
  size_t pb = (size_t)(base + t) * 64 + n;
  float dl = dlt[pd], uv = uu[pd], Bn = dbc[pb + 32];
  float h = 0.f, ssum = 0.f;

  for (int i = 0; i < CLEN - 1; ++i) {
    const int t2 = t + step;
    const size_t pd2 = (size_t)(base + t2) * DINNER + d;
    const size_t pb2 = (size_t)(base + t2) * 64 + n;
    const float dl2 = dlt[pd2];
    const float uv2 = uu[pd2];
    const float Bn2 = dbc[pb2 + 32];

    h = __expf(dl * An) * h + dl * Bn * uv;
    ssum += dl;
    t = t2; dl = dl2; uv = uv2; Bn = Bn2;
  }
  h = __expf(dl * An) * h + dl * Bn * uv;
  ssum += dl;

  Lbuf[(size_t)task * DSTATE + n] = h;
  Ebuf[(size_t)task * DSTATE + n] = __expf(An * ssum);
}

__global__ void scan_combine(const float* __restrict__ Lbuf,
                             const float* __restrict__ Ebuf,
                             float* __restrict__ Hin) {        // [NTASK,16]
  const int gid = blockIdx.x * blockDim.x + threadIdx.x;       // NCH*16
  const int n   = gid & 15;
  const int ch  = gid >> 4;
  float H = 0.f;
#pragma unroll
  for (int ck = 0; ck < NCHUNK; ++ck) {
    const size_t idx = (size_t)(ch * NCHUNK + ck) * DSTATE + n;
    Hin[idx] = H;
    H = Ebuf[idx] * H + Lbuf[idx];
  }
}

__global__ void scan_part2(const float* __restrict__ delta_f,
                           const float* __restrict__ delta_b,
                           const float* __restrict__ u_f,
                           const float* __restrict__ u_b,
                           const float* __restrict__ dbc_f,
                           const float* __restrict__ dbc_b,
                           const float* __restrict__ A_log,
                           const float* __restrict__ Dvec,
                           const float* __restrict__ Hin,
                           float* __restrict__ y_f,
                           float* __restrict__ y_b) {
  const int wave = threadIdx.x >> 5;
  const int lane = threadIdx.x & 31;
  const int n    = lane & 15;
  const int task = blockIdx.x * 16 + wave * 2 + (lane >> 4);
  const int ck   = task & (NCHUNK - 1);
  const int ch   = task >> 3;
  const int dir  = ch >> 10;
  const int bb   = (ch >> 9) & 1;
  const int d    = ch & (DINNER - 1);

  const float* dlt = dir ? delta_b : delta_f;
  const float* uu  = dir ? u_b : u_f;
  const float* dbc = dir ? dbc_b : dbc_f;
  float* y         = dir ? y_b : y_f;

  const float An = -__expf(A_log[d * DSTATE + n]);
  const float Dd = Dvec[d];
  const int base = bb * SEQ;
  const int step = dir ? -1 : 1;
  int t = dir ? (SEQ - 1 - ck * CLEN) : (ck * CLEN);

  size_t pd = (size_t)(base + t) * DINNER + d;
  size_t pb = (size_t)(base + t) * 64 + n;
  float dl = dlt[pd], uv = uu[pd], Bn = dbc[pb + 32], Cn = dbc[pb + 48];
  float h = Hin[(size_t)task * DSTATE + n];

  for (int i = 0; i < CLEN - 1; ++i) {
    const int t2 = t + step;
    const size_t pd2 = (size_t)(base + t2) * DINNER + d;
    const size_t pb2 = (size_t)(base + t2) * 64 + n;
    const float dl2 = dlt[pd2];
    const float uv2 = uu[pd2];
    const float Bn2 = dbc[pb2 + 32];
    const float Cn2 = dbc[pb2 + 48];

    h = __expf(dl * An) * h + dl * Bn * uv;
    float p = h * Cn;
    p += __shfl_xor(p, 8, 32);
    p += __shfl_xor(p, 4, 32);
    p += __shfl_xor(p, 2, 32);
    p += __shfl_xor(p, 1, 32);
    if (n == 0) y[(size_t)(base + t) * DINNER + d] = p + uv * Dd;

    t = t2; dl = dl2; uv = uv2; Bn = Bn2; Cn = Cn2;
  }
  h = __expf(dl * An) * h + dl * Bn * uv;
  float p = h * Cn;
  p += __shfl_xor(p, 8, 32);
  p += __shfl_xor(p, 4, 32);
  p += __shfl_xor(p, 2, 32);
  p += __shfl_xor(p, 1, 32);
  if (n == 0) y[(size_t)(base + t) * DINNER + d] = p + uv * Dd;
}

// ---------------------------------------------------------------------------
// 5) Gate + residual: out = (y_f + y_b) * silu(z) + x ; z = zx[:,512:1024]
// ---------------------------------------------------------------------------
__global__ void gate_kernel(const float* __restrict__ x,
                            const float* __restrict__ zx,
                            const float* __restrict__ y_f,
                            const float* __restrict__ y_b,
                            float* __restrict__ out) {
  const int gid = blockIdx.x * blockDim.x + threadIdx.x;
  const int d   = gid & (DINNER - 1);
  const int row = gid >> 9;
  const float z  = zx[(size_t)row * 1024 + 512 + d];
  const float zg = z / (1.f + __expf(-z));          // silu(z)
  out[(size_t)row * DINNER + d] =
      (y_f[(size_t)row * DINNER + d] + y_b[(size_t)row * DINNER + d]) * zg +
      x[(size_t)row * DINNER + d];
}

// ---------------------------------------------------------------------------
// Launcher. Workspace layout (floats), ~158 MB total.
// ---------------------------------------------------------------------------
extern "C" void kernel_launch(void* const* d_in, const int* in_sizes, int n_in,
                              void* d_out, int out_size, void* d_ws, size_t ws_size,
                              hipStream_t stream) {
  const float* x      = (const float*)d_in[0];
  const float* ln_g   = (const float*)d_in[1];
  const float* ln_b   = (const float*)d_in[2];
  const float* W_proj = (const float*)d_in[3];
  const float* conv_w = (const float*)d_in[4];
  const float* conv_b = (const float*)d_in[5];
  const float* W_x    = (const float*)d_in[6];
  const float* W_dt   = (const float*)d_in[7];
  const float* b_dt   = (const float*)d_in[8];
  const float* A_log  = (const float*)d_in[9];
  const float* Dvec   = (const float*)d_in[10];
  float* out = (float*)d_out;

  float* ws = (float*)d_ws;
  float* xn      = ws;                      ws += (size_t)ROWS * DIM;
  float* zx      = ws;                      ws += (size_t)ROWS * 1024;
  float* u_f     = ws;                      ws += (size_t)ROWS * DINNER;
  float* u_b     = ws;                      ws += (size_t)ROWS * DINNER;
  float* dbc_f   = ws;                      ws += (size_t)ROWS * 64;
  float* dbc_b   = ws;                      ws += (size_t)ROWS * 64;
  float* delta_f = ws;                      ws += (size_t)ROWS * DINNER;
  float* delta_b = ws;                      ws += (size_t)ROWS * DINNER;
  float* y_f     = ws;                      ws += (size_t)ROWS * DINNER;
  float* y_b     = ws;                      ws += (size_t)ROWS * DINNER;
  float* Lbuf    = ws;                      ws += (size_t)NTASK * DSTATE;
  float* Ebuf    = ws;                      ws += (size_t)NTASK * DSTATE;
  float* Hin     = ws;                      ws += (size_t)NTASK * DSTATE;

  // 1) LayerNorm
  ln_kernel<<<ROWS / 8, 256, 0, stream>>>(x, ln_g, ln_b, xn);

  // 2) zx = xn @ W_proj   (8192x1024, K=512)
  wmma_gemm_f32<0><<<dim3(ROWS / 128, 1024 / 64), 256, 0, stream>>>(
      xn, DIM, W_proj, 1024, zx, 1024, DIM, nullptr);

  // 3) causal/anticausal depthwise conv + softplus
  conv_kernel<<<(ROWS * DINNER) / 256, 256, 0, stream>>>(zx, conv_w, conv_b, u_f, u_b);

  // 4) dbc = u @ W_x      (8192x64, K=512), both directions
  wmma_gemm_f32<0><<<dim3(ROWS / 128, 1), 256, 0, stream>>>(
      u_f, DINNER, W_x, 64, dbc_f, 64, DINNER, nullptr);
  wmma_gemm_f32<0><<<dim3(ROWS / 128, 1), 256, 0, stream>>>(
      u_b, DINNER, W_x, 64, dbc_b, 64, DINNER, nullptr);

  // 5) delta = softplus(dt @ W_dt + b_dt)  (8192x512, K=32; dt = dbc[:, :32])
  wmma_gemm_f32<1><<<dim3(ROWS / 128, 512 / 64), 256, 0, stream>>>(
      dbc_f, 64, W_dt, 512, delta_f, DINNER, DTRANK, b_dt);
  wmma_gemm_f32<1><<<dim3(ROWS / 128, 512 / 64), 256, 0, stream>>>(
      dbc_b, 64, W_dt, 512, delta_b, DINNER, DTRANK, b_dt);

  // 6) chunked selective scan: local -> combine -> final
  scan_part1<<<NTASK / 16, 256, 0, stream>>>(delta_f, delta_b, u_f, u_b,
                                             dbc_f, dbc_b, A_log, Lbuf, Ebuf);
  scan_combine<<<(NCH * DSTATE) / 256, 256, 0, stream>>>(Lbuf, Ebuf, Hin);
  scan_part2<<<NTASK / 16, 256, 0, stream>>>(delta_f, delta_b, u_f, u_b,
                                             dbc_f, dbc_b, A_log, Dvec, Hin,
                                             y_f, y_b);

  // 7) gate + residual
  gate_kernel<<<(ROWS * DINNER) / 256, 256, 0, stream>>>(x, zx, y_f, y_b, out);
}